// GCNNet_3118146257467
// MI455X (gfx1250) — compile-verified
//
#include <hip/hip_runtime.h>

typedef __attribute__((ext_vector_type(2))) float v2f;
typedef __attribute__((ext_vector_type(8))) float v8f;

#define FEAT 128

// ---------------------------------------------------------------------------
// Elementwise helpers
// ---------------------------------------------------------------------------
__global__ void k_fill(float* __restrict__ p, float v, long long n) {
    long long i = (long long)blockIdx.x * blockDim.x + threadIdx.x;
    if (i < n) p[i] = v;
}

__global__ void k_deg(const int* __restrict__ dst, float* __restrict__ deg, long long E) {
    long long e = (long long)blockIdx.x * blockDim.x + threadIdx.x;
    if (e < E) atomicAdd(&deg[dst[e]], 1.0f);
}

__global__ void k_rsqrt_inplace(float* __restrict__ p, int n) {
    int i = blockIdx.x * blockDim.x + threadIdx.x;
    if (i < n) p[i] = rsqrtf(p[i]);   // deg >= 1 always (self loop)
}

// ---------------------------------------------------------------------------
// GEMM: O[N x 128] = H[N x 128] @ W[128 x 128] using V_WMMA_F32_16X16X4_F32.
// One wave per 16-row strip; 8 accumulator tiles cover all 128 output cols,
// so each A element is loaded exactly once. K-loop steps by 4.
//
// f32 16x16x4 operand layout (ISA 7.12.2):
//   A (16x4):  lanes 0-15 rows M=0..15 {K=k,k+1}; lanes 16-31 {K=k+2,k+3}
//   B (4x16):  lanes 0-15 cols N=0..15 {K=k,k+1}; lanes 16-31 {K=k+2,k+3}
//   C/D (16x16): vgpr r -> row r (lanes 0-15) / row r+8 (lanes 16-31)
// ---------------------------------------------------------------------------
__global__ __launch_bounds__(64) void k_gemm_wmma_f32(
    const float* __restrict__ H, const float* __restrict__ W,
    float* __restrict__ O, int N)
{
    const int wave = blockIdx.x * 2 + (threadIdx.x >> 5);
    const int lane = threadIdx.x & 31;
    const int m0   = wave * 16;
    if (m0 >= N) return;                       // wave-uniform exit (EXEC stays full)

    const int hi = (lane >= 16) ? 1 : 0;
    const int lr = lane & 15;

    int arow = m0 + lr;
    if (arow >= N) arow = N - 1;               // clamp loads on (non-existent) tail
    const float* __restrict__ A = H + (size_t)arow * FEAT;

    v8f acc[8] = {};

    for (int k = 0; k < FEAT; k += 4) {
        const int klo = k + (hi ? 2 : 0);      // even -> 8B aligned
        v2f a = *(const v2f*)(A + klo);
        const float* __restrict__ b0 = W + (size_t)klo * FEAT;
        const float* __restrict__ b1 = b0 + FEAT;
#pragma unroll
        for (int t = 0; t < 8; ++t) {
            v2f b;
            b.x = b0[t * 16 + lr];
            b.y = b1[t * 16 + lr];
            acc[t] = __builtin_amdgcn_wmma_f32_16x16x4_f32(
                /*neg_a=*/false, a, /*neg_b=*/false, b,
                /*c_mod=*/(short)0, acc[t],
                /*reuse_a=*/false, /*reuse_b=*/false);
        }
    }

#pragma unroll
    for (int t = 0; t < 8; ++t) {
#pragma unroll
        for (int r = 0; r < 8; ++r) {
            const int m = m0 + r + (hi ? 8 : 0);
            if (m < N) O[(size_t)m * FEAT + t * 16 + lr] = acc[t][r];
        }
    }
}

// ---------------------------------------------------------------------------
// Edge scatter: one wave per edge, 4 feats per lane.
// agg[col] += hw[row] * dinv[row]*dinv[col]
// ---------------------------------------------------------------------------
__global__ __launch_bounds__(256) void k_scatter(
    const float* __restrict__ hw,
    const int* __restrict__ src, const int* __restrict__ dst,
    const float* __restrict__ dinv,
    float* __restrict__ agg, long long E)
{
    long long t = (long long)blockIdx.x * blockDim.x + threadIdx.x;
    long long e = t >> 5;
    if (e >= E) return;
    const int lane = (int)(t & 31);
    const int r = src[e];
    const int c = dst[e];
    const float nrm = dinv[r] * dinv[c];
    float4 v = *(const float4*)(hw + (size_t)r * FEAT + lane * 4);
    float* d = agg + (size_t)c * FEAT + lane * 4;
    atomicAdd(d + 0, v.x * nrm);
    atomicAdd(d + 1, v.y * nrm);
    atomicAdd(d + 2, v.z * nrm);
    atomicAdd(d + 3, v.w * nrm);
}

// ---------------------------------------------------------------------------
// Epilogue: self-loop contribution + bias + ReLU
// h[i][f] = relu(agg[i][f] + hw[i][f]*dinv[i]^2 + bias[f])
// ---------------------------------------------------------------------------
__global__ __launch_bounds__(256) void k_bias_relu(
    const float* __restrict__ hw, const float* __restrict__ agg,
    const float* __restrict__ dinv, const float* __restrict__ bias,
    float* __restrict__ out, long long total)
{
    long long t = (long long)blockIdx.x * blockDim.x + threadIdx.x;
    if (t >= total) return;
    const int i = (int)(t >> 7);
    const int f = (int)(t & (FEAT - 1));
    const float d = dinv[i];
    float v = agg[t] + hw[t] * d * d + bias[f];
    out[t] = v > 0.0f ? v : 0.0f;
}

// ---------------------------------------------------------------------------
// Pool: batch is sorted, so a block scans a chunk of nodes keeping a register
// accumulator per feature and flushes one atomic per segment change.
// ---------------------------------------------------------------------------
__global__ __launch_bounds__(128) void k_pool_sum(
    const float* __restrict__ h, const int* __restrict__ batch,
    float* __restrict__ g, int N, int chunk)
{
    const int f = threadIdx.x;                 // 0..127
    const int start = blockIdx.x * chunk;
    if (start >= N) return;
    int end = start + chunk; if (end > N) end = N;

    int cur = batch[start];
    float sum = 0.0f;
    for (int n = start; n < end; ++n) {
        const int b = batch[n];
        if (b != cur) {
            atomicAdd(&g[(size_t)cur * FEAT + f], sum);
            sum = 0.0f;
            cur = b;
        }
        sum += h[(size_t)n * FEAT + f];
    }
    atomicAdd(&g[(size_t)cur * FEAT + f], sum);
}

__global__ void k_cnt(const int* __restrict__ batch, float* __restrict__ cnt, int N) {
    int i = blockIdx.x * blockDim.x + threadIdx.x;
    if (i < N) atomicAdd(&cnt[batch[i]], 1.0f);
}

// ---------------------------------------------------------------------------
// MLP head: one block per graph. g_mean -> Linear(128) -> ReLU -> Linear(1)
// ---------------------------------------------------------------------------
__global__ __launch_bounds__(128) void k_mlp_head(
    const float* __restrict__ g, const float* __restrict__ cnt,
    const float* __restrict__ W1, const float* __restrict__ b1,
    const float* __restrict__ W2, const float* __restrict__ b2,
    float* __restrict__ out)
{
    __shared__ float row[FEAT];
    __shared__ float z[FEAT];
    const int b = blockIdx.x;
    const int f = threadIdx.x;

    float c = cnt[b]; c = c > 1.0f ? c : 1.0f;
    row[f] = g[(size_t)b * FEAT + f] / c;
    __syncthreads();

    float acc = b1[f];
#pragma unroll 8
    for (int k = 0; k < FEAT; ++k) acc += row[k] * W1[(size_t)k * FEAT + f];
    acc = acc > 0.0f ? acc : 0.0f;
    z[f] = acc * W2[f];
    __syncthreads();

    for (int s = FEAT / 2; s > 0; s >>= 1) {
        if (f < s) z[f] += z[f + s];
        __syncthreads();
    }
    if (f == 0) out[b] = z[0] + b2[0];
}

// ---------------------------------------------------------------------------
// Orchestration
// ---------------------------------------------------------------------------
extern "C" void kernel_launch(void* const* d_in, const int* in_sizes, int n_in,
                              void* d_out, int out_size, void* d_ws, size_t ws_size,
                              hipStream_t stream) {
    const float* x     = (const float*)d_in[0];
    const int*   eidx  = (const int*)d_in[1];   // [2, E] int32
    const int*   batch = (const int*)d_in[2];   // [N]    int32 (sorted)
    const float* convW = (const float*)d_in[3]; // [L,128,128]
    const float* convB = (const float*)d_in[4]; // [L,128]
    const float* W1    = (const float*)d_in[5];
    const float* B1    = (const float*)d_in[6];
    const float* W2    = (const float*)d_in[7];
    const float* B2    = (const float*)d_in[8];
    float* out = (float*)d_out;

    const int       N = in_sizes[0] / FEAT;
    const long long E = in_sizes[1] / 2;
    const int       L = in_sizes[3] / (FEAT * FEAT);
    const int       G = out_size;

    const size_t nh = (size_t)N * FEAT;
    float* hbuf = (float*)d_ws;          // N*128
    float* hw   = hbuf + nh;             // N*128
    float* agg  = hw + nh;               // N*128
    float* dinv = agg + nh;              // N   (deg -> dinv in place)
    float* gbuf = dinv + N;              // G*128
    float* cnt  = gbuf + (size_t)G * FEAT; // G

    const int* src = eidx;               // edge_index[0] (source / row)
    const int* dst = eidx + E;           // edge_index[1] (target / col)

    // --- degree -> dinv (self loop contributes the initial 1.0) ---
    k_fill<<<(N + 255) / 256, 256, 0, stream>>>(dinv, 1.0f, N);
    k_deg<<<(int)((E + 255) / 256), 256, 0, stream>>>(dst, dinv, E);
    k_rsqrt_inplace<<<(N + 255) / 256, 256, 0, stream>>>(dinv, N);

    // --- zero pooling buffers once per launch ---
    hipMemsetAsync(gbuf, 0, ((size_t)G * FEAT + G) * sizeof(float), stream);

    const int gemm_tiles  = (N + 15) / 16;
    const int gemm_blocks = (gemm_tiles + 1) / 2;          // 2 waves / block
    const long long tot   = (long long)N * FEAT;
    const long long sthr  = E * 32;

    const float* hin = x;
    for (int l = 0; l < L; ++l) {
        k_gemm_wmma_f32<<<gemm_blocks, 64, 0, stream>>>(hin, convW + (size_t)l * FEAT * FEAT, hw, N);
        hipMemsetAsync(agg, 0, nh * sizeof(float), stream);
        k_scatter<<<(int)((sthr + 255) / 256), 256, 0, stream>>>(hw, src, dst, dinv, agg, E);
        k_bias_relu<<<(int)((tot + 255) / 256), 256, 0, stream>>>(hw, agg, dinv,
                                                                  convB + (size_t)l * FEAT, hbuf, tot);
        hin = hbuf;
    }

    // --- global mean pool ---
    const int chunk = 256;
    k_cnt<<<(N + 255) / 256, 256, 0, stream>>>(batch, cnt, N);
    k_pool_sum<<<(N + chunk - 1) / chunk, 128, 0, stream>>>(hbuf, batch, gbuf, N, chunk);

    // --- MLP head ---
    k_mlp_head<<<G, 128, 0, stream>>>(gbuf, cnt, W1, B1, W2, B2, out);
}